// MeanAggregator_46024869544579
// MI455X (gfx1250) — compile-verified
//
#include <hip/hip_runtime.h>
#include <hip/hip_bf16.h>
#include <stdint.h>

// MeanAggregator: out[b, :] = mean_k features[neigh_idx[b,k], :]
//   B = 16384 rows, K = 15 neighbors, D = 256 f32 features.
// Memory-bound gather+reduce; feature table (~20.5 MB) is L2-resident on
// MI455X (192 MB L2). Strategy: per-row wave32 issues the whole gather as
// CDNA5 async global->LDS b128 loads, then consumes rows from LDS pipelined
// against ASYNCcnt (async loads complete in order). Indices are preloaded in
// a batch so the 30 async gathers issue back-to-back instead of each being
// gated on an SMEM round trip.

#define AGG_K 15
#define AGG_D 256

typedef __attribute__((ext_vector_type(4))) float v4f;  // native vector: valid
                                                        // for nontemporal builtins

__global__ __launch_bounds__(32) void MeanAggregator_46024869544579_kernel(
    const int* __restrict__ neigh,     // [B, K] int32 neighbor ids
    const float* __restrict__ feat,    // [U, D] f32 features
    float* __restrict__ out)           // [B, D] f32 output
{
    __shared__ float smem[AGG_K * AGG_D];           // 15 KB staging: one 1KB row per neighbor

    const int      tid       = threadIdx.x;         // 0..31 (wave32)
    const int      row       = blockIdx.x;
    const int*     nb        = neigh + (size_t)row * AGG_K;
    const uint32_t lane_byte = (uint32_t)tid * 16u; // 16B (v4f) per lane per half-row

    // LDS byte offset of the staging buffer (flat ptr low 32 bits == LDS offset).
    const uint32_t lds_base = (uint32_t)(uintptr_t)&smem[0];

    // ---- Batch-preload all neighbor ids (workgroup-uniform -> wide s_load +
    // a single kmcnt wait) so the async issue loop below is not serialized on
    // per-index SMEM round trips.
    int idx[AGG_K];
#pragma unroll
    for (int k = 0; k < AGG_K; ++k) idx[k] = nb[k];

    // ---- Issue phase: 2 async b128 gathers per neighbor row (512B each, whole
    // wave covers 1KB with the offset:512 twin; the instruction offset applies
    // to BOTH the global and the LDS address, per the CDNA5 async pseudocode).
#pragma unroll
    for (int k = 0; k < AGG_K; ++k) {
        const uint64_t gaddr = (uint64_t)(uintptr_t)(feat + (size_t)idx[k] * AGG_D) + lane_byte;
        const uint32_t laddr = lds_base + (uint32_t)(k * AGG_D * 4) + lane_byte;
        asm volatile(
            "global_load_async_to_lds_b128 %0, %1, off\n\t"
            "global_load_async_to_lds_b128 %0, %1, off offset:512"
            :
            : "v"(laddr), "v"(gaddr)
            : "memory");
    }

    // ---- Consume phase: pipelined on ASYNCcnt. Async loads to global memory
    // complete in issue order, so after waiting ASYNCcnt <= 2*(K-1-k) the
    // loads for row k have landed in LDS while later rows are still in flight.
    v4f acc0 = (v4f)(0.0f);
    v4f acc1 = (v4f)(0.0f);
#pragma unroll
    for (int k = 0; k < AGG_K; ++k) {
        asm volatile("s_wait_asynccnt %0" :: "n"(2 * (AGG_K - 1 - k)) : "memory");
        const v4f* r = (const v4f*)&smem[k * AGG_D];
        acc0 += r[tid];        // floats [4*tid, 4*tid+4)
        acc1 += r[tid + 32];   // floats [128+4*tid, ...)
    }

    const float s = 1.0f / (float)AGG_K;
    acc0 *= s;
    acc1 *= s;

    // Streaming output (written once, ~17MB): non-temporal so it does not
    // evict the hot L2-resident feature table.
    v4f* o = (v4f*)(out + (size_t)row * AGG_D);
    __builtin_nontemporal_store(acc0, &o[tid]);
    __builtin_nontemporal_store(acc1, &o[tid + 32]);
}

extern "C" void kernel_launch(void* const* d_in, const int* in_sizes, int n_in,
                              void* d_out, int out_size, void* d_ws, size_t ws_size,
                              hipStream_t stream) {
    (void)n_in; (void)d_ws; (void)ws_size; (void)out_size;

    const int*   neigh = (const int*)d_in[0];    // [B, K] int32
    const float* feat  = (const float*)d_in[1];  // [U, D] f32
    float*       out   = (float*)d_out;          // [B, D] f32

    const int b = in_sizes[0] / AGG_K;           // number of output rows
    if (b <= 0) return;

    MeanAggregator_46024869544579_kernel<<<dim3((unsigned)b), dim3(32), 0, stream>>>(
        neigh, feat, out);
}